// Expansion_55044300865687
// MI455X (gfx1250) — compile-verified
//
#include <hip/hip_runtime.h>
#include <hip/hip_bf16.h>

typedef __attribute__((ext_vector_type(2))) float v2f;
typedef __attribute__((ext_vector_type(8))) float v8f;

// ---------------------------------------------------------------------------
// Constant-matrix builder: computes W[28][208] with W[k][n] = M[n][k],
// where Y[b, n] = sum_k M[n][k] * S[b][k]  (n = row*14+col of the 14x14 out).
// Runs entirely on device each launch (graph-capture safe, deterministic).
// ---------------------------------------------------------------------------

__device__ double dfact(int n) {
    double r = 1.0;
    for (int i = 2; i <= n; ++i) r *= (double)i;
    return r;
}

__device__ double su2_cg(int j1, int m1, int j2, int m2, int j3, int m3) {
    if (m3 != m1 + m2) return 0.0;
    int vmin = -j1 + j2 + m3;
    if (-j1 + m1 > vmin) vmin = -j1 + m1;
    if (0 > vmin) vmin = 0;
    int vmax = j2 + j3 + m1;
    if (j3 - j1 + j2 < vmax) vmax = j3 - j1 + j2;
    if (j3 + m3 < vmax) vmax = j3 + m3;
    double num = dfact(j3 + j1 - j2) * dfact(j3 - j1 + j2) * dfact(j1 + j2 - j3) *
                 dfact(j3 + m3) * dfact(j3 - m3);
    double den = dfact(j1 + j2 + j3 + 1) * dfact(j1 - m1) * dfact(j1 + m1) *
                 dfact(j2 - m2) * dfact(j2 + m2);
    double C = sqrt((2.0 * j3 + 1.0) * num / den);
    double S = 0.0;
    for (int v = vmin; v <= vmax; ++v) {
        double sgn = ((v + j2 + m2) & 1) ? -1.0 : 1.0;
        S += sgn * dfact(j2 + j3 + m1 - v) * dfact(j1 - m1 + v) /
             (dfact(v) * dfact(j3 - j1 + j2 - v) * dfact(j3 + m3 - v) *
              dfact(v + j1 - j2 - m3));
    }
    return C * S;
}

// q(l): complex change-of-basis real->complex SH (e3nn convention), d <= 9
__device__ void build_q(int l, double* qre, double* qim) {
    int d = 2 * l + 1;
    for (int i = 0; i < d * d; ++i) { qre[i] = 0.0; qim[i] = 0.0; }
    double s = sqrt(0.5);
    for (int m = -l; m < 0; ++m) {
        qre[(l + m) * d + (l - m)] = s;     // q[l+m, l+|m|] = 1/sqrt2
        qim[(l + m) * d + (l + m)] = -s;    // q[l+m, l-|m|] = -i/sqrt2
    }
    qre[l * d + l] = 1.0;
    for (int m = 1; m <= l; ++m) {
        double sg = (m & 1) ? -1.0 : 1.0;
        qre[(l + m) * d + (l + m)] = sg * s;  // (-1)^m / sqrt2
        qim[(l + m) * d + (l - m)] = sg * s;  // i (-1)^m / sqrt2
    }
    // multiply whole matrix by (-i)^l
    double pre, pim;
    switch (l & 3) {
        case 0: pre = 1.0;  pim = 0.0;  break;
        case 1: pre = 0.0;  pim = -1.0; break;
        case 2: pre = -1.0; pim = 0.0;  break;
        default: pre = 0.0; pim = 1.0;  break;
    }
    for (int i = 0; i < d * d; ++i) {
        double re = qre[i], im = qim[i];
        qre[i] = re * pre - im * pim;
        qim[i] = re * pim + im * pre;
    }
}

// real Wigner-3j, Frobenius-normalized; T[(a*d2+c)*d3+m], d1,d2<=5, d3<=9
__device__ void dev_w3j(int l1, int l2, int l3, double* T) {
    int d1 = 2 * l1 + 1, d2 = 2 * l2 + 1, d3 = 2 * l3 + 1;
    double C[5 * 5 * 9];
    for (int a = 0; a < d1; ++a)
        for (int b = 0; b < d2; ++b)
            for (int c = 0; c < d3; ++c)
                C[(a * d2 + b) * d3 + c] = su2_cg(l1, a - l1, l2, b - l2, l3, c - l3);
    double q1re[25], q1im[25], q2re[25], q2im[25], q3re[81], q3im[81];
    build_q(l1, q1re, q1im);
    build_q(l2, q2re, q2im);
    build_q(l3, q3re, q3im);
    double nrm = 0.0;
    for (int j = 0; j < d1; ++j)
        for (int l = 0; l < d2; ++l)
            for (int m = 0; m < d3; ++m) {
                double acc = 0.0;  // real part of complex triple product sum
                for (int i = 0; i < d1; ++i)
                    for (int k = 0; k < d2; ++k)
                        for (int n = 0; n < d3; ++n) {
                            double c = C[(i * d2 + k) * d3 + n];
                            if (c == 0.0) continue;
                            double are = q1re[i * d1 + j], aim = q1im[i * d1 + j];
                            double bre = q2re[k * d2 + l], bim = q2im[k * d2 + l];
                            double tre = are * bre - aim * bim;
                            double tim = are * bim + aim * bre;
                            // * conj(q3[n][m])
                            double cre = q3re[n * d3 + m], cim = q3im[n * d3 + m];
                            double ure = tre * cre + tim * cim;
                            acc += ure * c;
                        }
                T[(j * d2 + l) * d3 + m] = acc;
                nrm += acc * acc;
            }
    nrm = sqrt(nrm);
    double inv = (nrm > 0.0) ? 1.0 / nrm : 0.0;
    for (int t = 0; t < d1 * d2 * d3; ++t) T[t] *= inv;
}

__global__ void build_w_kernel(float* __restrict__ W) {
    // zero padded W[28][208]
    for (int idx = threadIdx.x; idx < 28 * 208; idx += blockDim.x) W[idx] = 0.0f;
    __syncthreads();
    if (threadIdx.x != 0) return;

    const int OFF25[5] = {0, 1, 4, 9, 16};   // offsets of s_i in 25-dim S
    const int LO[3]    = {0, 1, 2};          // l of OUT irreps
    const int UO[3]    = {3, 2, 1};          // mul of OUT irreps
    const int ROFF[3]  = {0, 3, 9};          // row/col block offsets in 14
    const int ins[14][3] = {
        {0,0,0},{0,1,1},{0,2,2},
        {1,0,1},{1,1,0},{1,1,2},{1,2,1},
        {2,0,2},{2,1,1},{2,2,0},{2,2,2},
        {3,1,2},{3,2,1},
        {4,2,2}};

    double T[5 * 5 * 9];
    for (int t = 0; t < 14; ++t) {
        int i = ins[t][0], j = ins[t][1], k = ins[t][2];
        int l1 = LO[j], l2 = LO[k], l3 = i;   // input irrep i has l = i
        int d1 = 2 * l1 + 1, d2 = 2 * l2 + 1, d3 = 2 * l3 + 1;
        dev_w3j(l1, l2, l3, T);
        int u = UO[j], v = UO[k];
        for (int uu = 0; uu < u; ++uu)
            for (int a = 0; a < d1; ++a)
                for (int vv = 0; vv < v; ++vv)
                    for (int c = 0; c < d2; ++c) {
                        int row = ROFF[j] + uu * d1 + a;
                        int col = ROFF[k] + vv * d2 + c;
                        int n = row * 14 + col;       // 0..195
                        for (int m = 0; m < d3; ++m)
                            W[(OFF25[i] + m) * 208 + n] += (float)T[(a * d2 + c) * d3 + m];
                    }
    }
}

// ---------------------------------------------------------------------------
// Stage 1: channel reduction. One wave32 per batch row; lane handles channels
// lane, lane+32, lane+64, lane+96 of each irrep; 25 accumulators; shuffle
// reduce across the wave; lane 0 writes S[b][0..27] (cols 25..27 zero-pad).
// Non-temporal loads: the 640 MB input stream is single-use.
// ---------------------------------------------------------------------------
__global__ void reduce_channels_kernel(const float* __restrict__ x,
                                       float* __restrict__ S, int B) {
    int wave = (int)((blockIdx.x * blockDim.x + threadIdx.x) >> 5);
    int lane = threadIdx.x & 31;
    if (wave >= B) return;
    const float* xb = x + (size_t)wave * 3200;

    float s[25];
#pragma unroll
    for (int i = 0; i < 25; ++i) s[i] = 0.0f;

#pragma unroll
    for (int cc = 0; cc < 4; ++cc) {
        int ch = lane + 32 * cc;
        s[0] += __builtin_nontemporal_load(xb + ch);      // l=0 @ 0
        {
            const float* p = xb + 128 + ch * 3;           // l=1 @ 128
#pragma unroll
            for (int e = 0; e < 3; ++e) s[1 + e] += __builtin_nontemporal_load(p + e);
        }
        {
            const float* p = xb + 512 + ch * 5;           // l=2 @ 512
#pragma unroll
            for (int e = 0; e < 5; ++e) s[4 + e] += __builtin_nontemporal_load(p + e);
        }
        {
            const float* p = xb + 1152 + ch * 7;          // l=3 @ 1152
#pragma unroll
            for (int e = 0; e < 7; ++e) s[9 + e] += __builtin_nontemporal_load(p + e);
        }
        {
            const float* p = xb + 2048 + ch * 9;          // l=4 @ 2048
#pragma unroll
            for (int e = 0; e < 9; ++e) s[16 + e] += __builtin_nontemporal_load(p + e);
        }
    }

#pragma unroll
    for (int off = 16; off > 0; off >>= 1) {
#pragma unroll
        for (int i = 0; i < 25; ++i) s[i] += __shfl_down(s[i], off, 32);
    }

    if (lane == 0) {
        float* Sp = S + (size_t)wave * 28;
#pragma unroll
        for (int i = 0; i < 25; ++i) Sp[i] = s[i];
        Sp[25] = 0.0f; Sp[26] = 0.0f; Sp[27] = 0.0f;
    }
}

// ---------------------------------------------------------------------------
// Stage 2: Y[B,196] = S[B,28] @ W[28,208] (padded), via f32 WMMA 16x16x4.
// One wave per 16(batch) x 16(outcol) tile; 7 K-steps of K=4.
// A layout (16x4 f32): lane<16 -> M=lane,K={k0,k0+1}; lane>=16 -> K={k0+2,k0+3}
// B layout (4x16 f32): lanes hold N=lane&15; lane-half + vgpr give K (mirrors A)
// C/D layout: VGPR r -> M = 8*(lane>>4) + r, N = lane&15
// Store epilogue: single base address + immediate offsets (rows stride 784 B);
// col-guard exec mask set once; partial row-groups take a wave-uniform slow path.
// ---------------------------------------------------------------------------
__global__ void expand_wmma_kernel(const float* __restrict__ S,
                                   const float* __restrict__ W,
                                   float* __restrict__ Y,
                                   int B, int nTilesTotal) {
    int waveId = (int)((blockIdx.x * blockDim.x + threadIdx.x) >> 5);
    if (waveId >= nTilesTotal) return;  // wave-uniform exit: EXEC full below
    int rg = waveId / 13;
    int nt = waveId - rg * 13;
    int m0 = rg * 16;
    int n0 = nt * 16;

    int lane = threadIdx.x & 31;
    int half = lane >> 4;
    int lm = lane & 15;

    int arow = m0 + lm;
    if (arow > B - 1) arow = B - 1;  // clamp (no EXEC divergence)
    const float* Sr = S + (size_t)arow * 28 + 2 * half;
    const float* Wc = W + (size_t)(2 * half) * 208 + n0 + lm;

    v8f acc = {0.f, 0.f, 0.f, 0.f, 0.f, 0.f, 0.f, 0.f};
#pragma unroll
    for (int k0 = 0; k0 < 28; k0 += 4) {
        v2f a;
        a.x = Sr[k0 + 0];
        a.y = Sr[k0 + 1];
        v2f b;
        b.x = Wc[(size_t)(k0 + 0) * 208];
        b.y = Wc[(size_t)(k0 + 1) * 208];
        acc = __builtin_amdgcn_wmma_f32_16x16x4_f32(
            /*neg_a=*/false, a, /*neg_b=*/false, b,
            /*c_mod=*/(short)0, acc, /*reuse_a=*/false, /*reuse_b=*/false);
    }

    int col = n0 + lm;
    int row0 = m0 + half * 8;                   // first of 8 consecutive rows
    float* yp = Y + (size_t)row0 * 196 + col;   // rows stride 196 floats

    if (m0 + 16 <= B) {
        // full row-group (always the case when B % 16 == 0): one exec-mask
        // update for the col guard, 8 stores off one base with imm offsets
        if (col < 196) {
#pragma unroll
            for (int r = 0; r < 8; ++r) yp[(size_t)r * 196] = acc[r];
        }
    } else {
        // partial last row-group (generic-B safety; wave-uniform branch)
        if (col < 196) {
#pragma unroll
            for (int r = 0; r < 8; ++r)
                if (row0 + r < B) yp[(size_t)r * 196] = acc[r];
        }
    }
}

// ---------------------------------------------------------------------------
extern "C" void kernel_launch(void* const* d_in, const int* in_sizes, int n_in,
                              void* d_out, int out_size, void* d_ws, size_t ws_size,
                              hipStream_t stream) {
    const float* x = (const float*)d_in[0];
    int B = in_sizes[0] / 3200;   // 50000

    float* W = (float*)d_ws;                          // 28*208 floats (~23 KB)
    float* S = (float*)((char*)d_ws + 32768);         // B*28 floats (~5.6 MB)
    float* Y = (float*)d_out;                         // B*196 floats

    // 1) build constant projection matrix on device (capture-safe)
    build_w_kernel<<<1, 64, 0, stream>>>(W);

    // 2) channel reduction: one wave per batch row, 8 waves per block
    {
        int blocks = (B + 7) / 8;
        reduce_channels_kernel<<<blocks, 256, 0, stream>>>(x, S, B);
    }

    // 3) WMMA projection: one wave per 16x16 tile, 8 waves per block
    {
        int rowGroups = (B + 15) / 16;
        int tiles = rowGroups * 13;          // 13 N-tiles cover 196 cols
        int blocks = (tiles + 7) / 8;
        expand_wmma_kernel<<<blocks, 256, 0, stream>>>(S, W, Y, B, tiles);
    }
}